// FourierBlock_35450660061459
// MI455X (gfx1250) — compile-verified
//
#include <hip/hip_runtime.h>
#include <math.h>

typedef __attribute__((ext_vector_type(2))) float v2f;
typedef __attribute__((ext_vector_type(8))) float v8f;

#define D_MODEL 1024
#define N_HEADS 16
#define D_HEAD  64
#define MODES   32
#define LSEQ    4096
#define NBATCH  4
#define MROWS   16384   // NBATCH * LSEQ

#define BM 64
#define BN 64
#define BK 32

// ---------------------------------------------------------------------------
// Generic tiled f32 WMMA GEMM:  C = A * op(B) (+ bias)
//   A: [M,K] row-major, leading dim lda
//   B: BT ? [N,K] row-major (i.e. computes A@B^T) : [K,N] row-major
//   C: [M,N] row-major, leading dim ldc
// Batch (grid.z): per-batch offsets = (z/zdiv)*s?1 + (z%zdiv)*s?2  (elements)
// ---------------------------------------------------------------------------
template <bool BT>
__global__ __launch_bounds__(256) void gemm_wmma_f32(
    const float* __restrict__ A, long long sA1, long long sA2, int lda,
    const float* __restrict__ B, long long sB1, long long sB2, int ldb,
    const float* __restrict__ bias,
    float* __restrict__ C, long long sC1, long long sC2, int ldc,
    int M, int N, int K, int zdiv)
{
    __shared__ float As[BM][BK + 1];
    __shared__ float Bs[BN][BK + 1];   // N-major: Bs[n][k]

    const int z  = blockIdx.z;
    const int zq = z / zdiv, zr = z % zdiv;
    A += (size_t)zq * sA1 + (size_t)zr * sA2;
    B += (size_t)zq * sB1 + (size_t)zr * sB2;
    C += (size_t)zq * sC1 + (size_t)zr * sC2;

    const int tileM = blockIdx.x * BM;
    const int tileN = blockIdx.y * BN;

    const int tid  = threadIdx.x;
    const int wave = tid >> 5;          // 0..7
    const int lane = tid & 31;
    const int wm   = wave >> 2;         // 0..1 : rows [wm*32, wm*32+32)
    const int wn   = wave & 3;          // 0..3 : cols [wn*16, wn*16+16)

    const int lrow = lane & 15;         // M (A) / N (B) index within frag
    const int lko  = (lane >> 4) << 1;  // K sub-offset: 0 (lanes 0-15) or 2

    v8f acc0 = {};
    v8f acc1 = {};

    for (int k0 = 0; k0 < K; k0 += BK) {
        // --- stage A tile: As[m][k] = A[(tileM+m)*lda + k0+k] (8 floats/thread)
        {
            const int m  = tid >> 2;
            const int kk = (tid & 3) << 3;
            const float4* src =
                reinterpret_cast<const float4*>(&A[(size_t)(tileM + m) * lda + k0 + kk]);
            float4 p0 = src[0], p1 = src[1];
            As[m][kk + 0] = p0.x; As[m][kk + 1] = p0.y;
            As[m][kk + 2] = p0.z; As[m][kk + 3] = p0.w;
            As[m][kk + 4] = p1.x; As[m][kk + 5] = p1.y;
            As[m][kk + 6] = p1.z; As[m][kk + 7] = p1.w;
        }
        // --- stage B tile into Bs[n][k]
        if constexpr (BT) {
            // B is [N,K]: contiguous along k
            const int n  = tid >> 2;
            const int kk = (tid & 3) << 3;
            const float4* src =
                reinterpret_cast<const float4*>(&B[(size_t)(tileN + n) * ldb + k0 + kk]);
            float4 p0 = src[0], p1 = src[1];
            Bs[n][kk + 0] = p0.x; Bs[n][kk + 1] = p0.y;
            Bs[n][kk + 2] = p0.z; Bs[n][kk + 3] = p0.w;
            Bs[n][kk + 4] = p1.x; Bs[n][kk + 5] = p1.y;
            Bs[n][kk + 6] = p1.z; Bs[n][kk + 7] = p1.w;
        } else {
            // B is [K,N]: contiguous along n
            const int k  = tid >> 3;           // 0..31
            const int n0 = (tid & 7) << 3;     // 0,8,...,56
            const float4* src =
                reinterpret_cast<const float4*>(&B[(size_t)(k0 + k) * ldb + tileN + n0]);
            float4 p0 = src[0], p1 = src[1];
            Bs[n0 + 0][k] = p0.x; Bs[n0 + 1][k] = p0.y;
            Bs[n0 + 2][k] = p0.z; Bs[n0 + 3][k] = p0.w;
            Bs[n0 + 4][k] = p1.x; Bs[n0 + 5][k] = p1.y;
            Bs[n0 + 6][k] = p1.z; Bs[n0 + 7][k] = p1.w;
        }
        __syncthreads();

        // --- 16x16x4 f32 WMMA over the 32-deep K slab
        #pragma unroll
        for (int kk = 0; kk < BK; kk += 4) {
            // B frag (4x16): lanes 0-15 hold K=kk..kk+1, lanes 16-31 K=kk+2..kk+3
            v2f b;
            b.x = Bs[wn * 16 + lrow][kk + lko];
            b.y = Bs[wn * 16 + lrow][kk + lko + 1];
            // A frags (16x4), same K split
            v2f a0, a1;
            a0.x = As[wm * 32 + lrow][kk + lko];
            a0.y = As[wm * 32 + lrow][kk + lko + 1];
            a1.x = As[wm * 32 + 16 + lrow][kk + lko];
            a1.y = As[wm * 32 + 16 + lrow][kk + lko + 1];
            acc0 = __builtin_amdgcn_wmma_f32_16x16x4_f32(
                false, a0, false, b, (short)0, acc0, false, false);
            acc1 = __builtin_amdgcn_wmma_f32_16x16x4_f32(
                false, a1, false, b, (short)0, acc1, false, false);
        }
        __syncthreads();
    }

    // --- epilogue: C/D layout — VGPR r: lanes 0-15 row r, lanes 16-31 row r+8
    const int col = lane & 15;
    const int rb  = (lane >> 4) << 3;   // 0 or 8
    const int gn  = tileN + wn * 16 + col;
    const float badd = bias ? bias[gn] : 0.0f;
    #pragma unroll
    for (int r = 0; r < 8; ++r) {
        const int gm0 = tileM + wm * 32 + rb + r;
        C[(size_t)gm0 * ldc + gn] = acc0[r] + badd;
        C[(size_t)(gm0 + 16) * ldc + gn] = acc1[r] + badd;
    }
}

// ---------------------------------------------------------------------------
// DFT bases. F: [2*MODES][L]  rows 2k=cos, 2k+1=-sin  (forward, truncated rfft)
//            G: [L][2*MODES]  cols 2k=s_k*cos, 2k+1=-s_k*sin (irfft of trunc.)
// s_0 = 1/L, s_k = 2/L (k>=1). Integer phase mod keeps sincos args in [0,2pi).
// ---------------------------------------------------------------------------
__global__ void build_basis_kernel(float* __restrict__ F, float* __restrict__ G)
{
    const int idx = blockIdx.x * blockDim.x + threadIdx.x;   // L*MODES
    if (idx >= LSEQ * MODES) return;
    const int k = idx & (MODES - 1);
    const int t = idx >> 5;
    const int phase = (k * t) & (LSEQ - 1);
    const float th = (float)phase * (6.28318530717958647692f / (float)LSEQ);
    float s, c;
    sincosf(th, &s, &c);
    F[(size_t)(2 * k) * LSEQ + t]     = c;
    F[(size_t)(2 * k + 1) * LSEQ + t] = -s;
    const float sc = (k == 0) ? (1.0f / (float)LSEQ) : (2.0f / (float)LSEQ);
    G[(size_t)t * (2 * MODES) + 2 * k]     = sc * c;
    G[(size_t)t * (2 * MODES) + 2 * k + 1] = -sc * s;
}

// ---------------------------------------------------------------------------
// Spectral pointwise complex multiply: Y = X * (wr + i*wi), per (b,h,k,d).
// X/Y layout: [b][h][2*MODES rows (Re/Im interleaved)][D_HEAD]
// ---------------------------------------------------------------------------
__global__ void spectral_mul_kernel(const float* __restrict__ X,
                                    const float* __restrict__ wr,
                                    const float* __restrict__ wi,
                                    float* __restrict__ Y)
{
    const int idx = blockIdx.x * blockDim.x + threadIdx.x;
    if (idx >= NBATCH * N_HEADS * MODES * D_HEAD) return;
    const int d = idx & 63;
    const int k = (idx >> 6) & 31;
    const int h = (idx >> 11) & 15;
    const int b = idx >> 15;
    const size_t base = (size_t)(b * N_HEADS + h) * (2 * MODES) * D_HEAD;
    const float xr = X[base + (size_t)(2 * k) * D_HEAD + d];
    const float xi = X[base + (size_t)(2 * k + 1) * D_HEAD + d];
    const size_t wofs = ((size_t)h * MODES + k) * D_HEAD + d;
    const float wre = wr[wofs];
    const float wim = wi[wofs];
    Y[base + (size_t)(2 * k) * D_HEAD + d]     = xr * wre - xi * wim;
    Y[base + (size_t)(2 * k + 1) * D_HEAD + d] = xr * wim + xi * wre;
}

// ---------------------------------------------------------------------------
extern "C" void kernel_launch(void* const* d_in, const int* in_sizes, int n_in,
                              void* d_out, int out_size, void* d_ws, size_t ws_size,
                              hipStream_t stream)
{
    (void)in_sizes; (void)n_in; (void)out_size; (void)ws_size;
    const float* q  = (const float*)d_in[0];
    const float* Wv = (const float*)d_in[1];
    const float* bv = (const float*)d_in[2];
    const float* Wo = (const float*)d_in[3];
    const float* bo = (const float*)d_in[4];
    const float* wr = (const float*)d_in[5];
    const float* wi = (const float*)d_in[6];
    float* out = (float*)d_out;

    char* ws = (char*)d_ws;
    float* F = (float*)(ws + (size_t)0 * (1 << 20));      // 1 MB  [64][4096]
    float* G = (float*)(ws + (size_t)1 * (1 << 20));      // 1 MB  [4096][64]
    float* X = (float*)(ws + (size_t)2 * (1 << 20));      // 1 MB  [64][64][64]
    float* Y = (float*)(ws + (size_t)3 * (1 << 20));      // 1 MB
    float* V = (float*)(ws + (size_t)4 * (1 << 20));      // 64 MB [16384][1024], reused as irfft out

    // 0) DFT bases
    build_basis_kernel<<<(LSEQ * MODES + 255) / 256, 256, 0, stream>>>(F, G);

    // 1) V = q @ Wv^T + bv    [16384,1024]x[1024,1024]^T
    dim3 gProj(MROWS / BM, D_MODEL / BN, 1);
    gemm_wmma_f32<true><<<gProj, 256, 0, stream>>>(
        q, 0, 0, D_MODEL,
        Wv, 0, 0, D_MODEL,
        bv,
        V, 0, 0, D_MODEL,
        MROWS, D_MODEL, D_MODEL, 1);

    // 2) Forward truncated DFT per (b,h): X = F[64,4096] @ V_bh[4096,64]
    dim3 gFwd(1, 1, NBATCH * N_HEADS);
    gemm_wmma_f32<false><<<gFwd, 256, 0, stream>>>(
        F, 0, 0, LSEQ,
        V, (long long)LSEQ * D_MODEL, D_HEAD, D_MODEL,
        nullptr,
        X, (long long)N_HEADS * 2 * MODES * D_HEAD, 2 * MODES * D_HEAD, D_HEAD,
        2 * MODES, D_HEAD, LSEQ, N_HEADS);

    // 3) Spectral complex multiply
    spectral_mul_kernel<<<(NBATCH * N_HEADS * MODES * D_HEAD + 255) / 256, 256, 0, stream>>>(
        X, wr, wi, Y);

    // 4) Inverse DFT per (b,h): yt_bh[4096,64] = G[4096,64] @ Y_bh[64,64]
    //    yt aliases V (V fully consumed by step 2).
    dim3 gInv(LSEQ / BM, 1, NBATCH * N_HEADS);
    gemm_wmma_f32<false><<<gInv, 256, 0, stream>>>(
        G, 0, 0, 2 * MODES,
        Y, (long long)N_HEADS * 2 * MODES * D_HEAD, 2 * MODES * D_HEAD, D_HEAD,
        nullptr,
        V, (long long)LSEQ * D_MODEL, D_HEAD, D_MODEL,
        LSEQ, D_HEAD, 2 * MODES, N_HEADS);

    // 5) out = yt @ Wo^T + bo
    gemm_wmma_f32<true><<<gProj, 256, 0, stream>>>(
        V, 0, 0, D_MODEL,
        Wo, 0, 0, D_MODEL,
        bo,
        out, 0, 0, D_MODEL,
        MROWS, D_MODEL, D_MODEL, 1);
}